// CentroidAwareVoxelization_36258113913322
// MI455X (gfx1250) — compile-verified
//
#include <hip/hip_runtime.h>
#include <math.h>

// ---------------------------------------------------------------------------
// CentroidAwareVoxelization for MI455X (gfx1250), wave32 + WMMA fp32 path.
//  - V_WMMA_F32_16X16X4_F32 GEMMs, W staged in LDS via async global->LDS.
//  - BN batch stats fused into GEMM epilogue (ds_add_f32 + global atomics).
//  - BN+GELU of previous layer fused into A-fragment load of the next GEMM.
//  - concat layer fully virtual: A fragments of GEMM3 assembled on the fly
//    from pts / gelu(bn2(Y2)) / voxel centroids (no concat buffer at all).
// ---------------------------------------------------------------------------

typedef float v2f __attribute__((ext_vector_type(2)));
typedef float v8f __attribute__((ext_vector_type(8)));
typedef int   a4i __attribute__((vector_size(16)));   // builtin's int4 type

#define PTS_B   4
#define PTS_N   65536
#define PTOT    (PTS_B * PTS_N)   // 262144 points
#define DD      128
#define TS      (1u << 19)        // hash table slots
#define TSMASK  (TS - 1u)
#define VOXEL   0.1f
#define BNEPS   1e-5f
#define CONCATK 136               // 134 padded to multiple of 4

#if defined(__has_builtin)
#if __has_builtin(__builtin_amdgcn_global_load_async_to_lds_b128)
#define HAVE_ASYNC_LDS 1
#endif
#if __has_builtin(__builtin_amdgcn_s_wait_asynccnt)
#define HAVE_WAIT_ASYNC 1
#endif
#endif

__device__ __forceinline__ float gelu_exact(float x) {
  return 0.5f * x * (1.0f + erff(x * 0.70710678118654752f));
}

// Cooperative stage of W (totalFloats, 16B-aligned) into LDS.
__device__ __forceinline__ void stage_w_to_lds(const float* __restrict__ W,
                                               float* sW, int totalFloats) {
#if defined(HAVE_ASYNC_LDS)
  for (int off = threadIdx.x * 4; off < totalFloats; off += 256 * 4) {
    __builtin_amdgcn_global_load_async_to_lds_b128(
        (__attribute__((address_space(1))) a4i*)(W + off),
        (__attribute__((address_space(3))) a4i*)(sW + off), 0, 0);
  }
#if defined(HAVE_WAIT_ASYNC)
  __builtin_amdgcn_s_wait_asynccnt(0);
#else
  asm volatile("s_wait_asynccnt 0x0" ::: "memory");
#endif
#else
  for (int off = threadIdx.x * 4; off < totalFloats; off += 256 * 4) {
    float4 v = *(const float4*)(W + off);
    *(float4*)(sW + off) = v;
  }
#endif
  __syncthreads();
}

// ---------------------------------------------------------------------------
// 1) Spatial-hash voxelization: claim unique voxel ids, accumulate centroids.
// ---------------------------------------------------------------------------
__global__ __launch_bounds__(256) void hash_build_kernel(
    const float* __restrict__ pts, int* __restrict__ keys, int* __restrict__ vals,
    int* __restrict__ counter, float* __restrict__ counts, float* __restrict__ csum,
    int* __restrict__ uvox, int* __restrict__ inv)
{
  int i = blockIdx.x * 256 + threadIdx.x;
  if (i >= PTOT) return;
  float px = pts[3 * i + 0], py = pts[3 * i + 1], pz = pts[3 * i + 2];
  int vx = (int)floorf(px / VOXEL);
  int vy = (int)floorf(py / VOXEL);
  int vz = (int)floorf(pz / VOXEL);
  unsigned h = (unsigned)vx * 73856093u + (unsigned)vy * 19349663u +
               (unsigned)vz * 83492791u;                       // int32-wrap hash
  unsigned slot = (h * 2654435761u) & TSMASK;
  int vid = -1;
  for (;;) {
    int prev = atomicCAS(&keys[slot], -1, (int)h);
    if (prev == -1) {                       // claimed a fresh voxel slot
      vid = atomicAdd(counter, 1);
      uvox[vid * 4 + 0] = i >> 16;          // batch index (N = 65536)
      uvox[vid * 4 + 1] = vx;
      uvox[vid * 4 + 2] = vy;
      uvox[vid * 4 + 3] = vz;
      __hip_atomic_store(&vals[slot], vid, __ATOMIC_RELEASE, __HIP_MEMORY_SCOPE_AGENT);
      break;
    }
    if (prev == (int)h) {                   // existing voxel: wait for publish
      int v = __hip_atomic_load(&vals[slot], __ATOMIC_ACQUIRE, __HIP_MEMORY_SCOPE_AGENT);
      while (v < 0)
        v = __hip_atomic_load(&vals[slot], __ATOMIC_ACQUIRE, __HIP_MEMORY_SCOPE_AGENT);
      vid = v;
      break;
    }
    slot = (slot + 1u) & TSMASK;            // linear probe
  }
  inv[i] = vid;
  atomicAdd(&counts[vid], 1.0f);
  atomicAdd(&csum[3 * vid + 0], px);
  atomicAdd(&csum[3 * vid + 1], py);
  atomicAdd(&csum[3 * vid + 2], pz);
}

__global__ __launch_bounds__(256) void centroid_finalize_kernel(
    const float* __restrict__ counts, const float* __restrict__ csum,
    float* __restrict__ centroid)
{
  int j = blockIdx.x * 256 + threadIdx.x;
  if (j >= PTOT) return;
  float d = fmaxf(counts[j], 1.0f);
  centroid[3 * j + 0] = csum[3 * j + 0] / d;
  centroid[3 * j + 1] = csum[3 * j + 1] / d;
  centroid[3 * j + 2] = csum[3 * j + 2] / d;
}

// norm_points output + X1 (K padded 3->4) + transposed uvox output
__global__ __launch_bounds__(256) void post_voxel_kernel(
    const float* __restrict__ pts, const int* __restrict__ inv,
    const float* __restrict__ centroid, const int* __restrict__ uvox,
    float* __restrict__ norm_out, float* __restrict__ x1, int* __restrict__ uvoxT)
{
  int i = blockIdx.x * 256 + threadIdx.x;
  if (i >= PTOT) return;
  int v = inv[i];
  float nx = pts[3 * i + 0] - centroid[3 * v + 0];
  float ny = pts[3 * i + 1] - centroid[3 * v + 1];
  float nz = pts[3 * i + 2] - centroid[3 * v + 2];
  norm_out[3 * i + 0] = nx;
  norm_out[3 * i + 1] = ny;
  norm_out[3 * i + 2] = nz;
  x1[4 * i + 0] = nx;
  x1[4 * i + 1] = ny;
  x1[4 * i + 2] = nz;
  x1[4 * i + 3] = 0.0f;
  uvoxT[0 * PTOT + i] = uvox[i * 4 + 0];
  uvoxT[1 * PTOT + i] = uvox[i * 4 + 1];
  uvoxT[2 * PTOT + i] = uvox[i * 4 + 2];
  uvoxT[3 * PTOT + i] = uvox[i * 4 + 3];
}

// ---------------------------------------------------------------------------
// 2) Weight padding (K must be a multiple of 4 for V_WMMA_F32_16X16X4_F32)
// ---------------------------------------------------------------------------
__global__ __launch_bounds__(256) void pack_w1_kernel(const float* __restrict__ w1,
                                                      float* __restrict__ w1p)
{
  int i = blockIdx.x * 256 + threadIdx.x;   // 4*128
  if (i >= 4 * DD) return;
  w1p[i] = (i < 3 * DD) ? w1[i] : 0.0f;
}

__global__ __launch_bounds__(256) void pack_w3_kernel(const float* __restrict__ w3,
                                                      float* __restrict__ w3p)
{
  int i = blockIdx.x * 256 + threadIdx.x;   // 136*128
  if (i >= CONCATK * DD) return;
  w3p[i] = (i < 134 * DD) ? w3[i] : 0.0f;
}

// ---------------------------------------------------------------------------
// 3) WMMA fp32 GEMM.  MODE 0: plain A from X (stride Kp)
//                     MODE 1: A = gelu(bn_prev(X)) (X = prev Y, stride 128)
//                     MODE 2: virtual concat A: [pts | gelu(bn2(Y2)) | cen | 0]
//    W staged in LDS via async global->LDS; B fragments served by ds_load.
//    Column sum / sumsq of Y accumulate via LDS ds_add_f32 + global atomics.
// ---------------------------------------------------------------------------
template <int MODE>
__global__ __launch_bounds__(256) void gemm_wmma_f32_kernel(
    const float* __restrict__ X, const float* __restrict__ inScale,
    const float* __restrict__ inShift, const float* __restrict__ pts,
    const float* __restrict__ cen, const float* __restrict__ W,
    float* __restrict__ Y, float* __restrict__ gsum, float* __restrict__ gsq,
    int Kp)
{
  __shared__ float sW[CONCATK * DD];        // staged weights (<= 68KB)
  __shared__ float lsum[DD], lsq[DD];
  __shared__ float sScale[DD], sShift[DD];

  for (int c = threadIdx.x; c < DD; c += 256) {
    lsum[c] = 0.0f;
    lsq[c] = 0.0f;
    if (MODE != 0) {
      sScale[c] = inScale[c];
      sShift[c] = inShift[c];
    }
  }
  stage_w_to_lds(W, sW, Kp * DD);           // includes __syncthreads()

  const int wave = threadIdx.x >> 5;
  const int lane = threadIdx.x & 31;
  const int rowBase = blockIdx.x * 128 + wave * 16;
  const int m = lane & 15;                 // M row within tile / N col in tile
  const int khalf = (lane >> 4) << 1;      // 0 for lanes 0-15, 2 for lanes 16-31
  const int row = rowBase + m;

  v8f acc[8];
#pragma unroll
  for (int t = 0; t < 8; ++t)
#pragma unroll
    for (int r = 0; r < 8; ++r) acc[t][r] = 0.0f;

  const float* xrow = X + (size_t)row * (MODE == 2 ? DD : Kp);

  for (int k = 0; k < Kp; k += 4) {
    if (k + 8 < (MODE == 2 ? DD : Kp))
      __builtin_prefetch(xrow + k + 8, 0, 1);                 // global_prefetch_b8
    float a0, a1;
    if (MODE == 2) {                        // virtual concat row fetch
      const int k0 = k + khalf, k1 = k + khalf + 1;
      // element 0
      if (k0 < 3)        a0 = pts[row * 3 + k0];
      else if (k0 < 131) a0 = gelu_exact(xrow[k0 - 3] * sScale[k0 - 3] + sShift[k0 - 3]);
      else if (k0 < 134) a0 = cen[row * 3 + (k0 - 131)];
      else               a0 = 0.0f;
      // element 1
      if (k1 < 3)        a1 = pts[row * 3 + k1];
      else if (k1 < 131) a1 = gelu_exact(xrow[k1 - 3] * sScale[k1 - 3] + sShift[k1 - 3]);
      else if (k1 < 134) a1 = cen[row * 3 + (k1 - 131)];
      else               a1 = 0.0f;
    } else {
      a0 = xrow[k + khalf + 0];             // A 16x4 fp32 layout (lane=M, K pair)
      a1 = xrow[k + khalf + 1];
      if (MODE == 1) {                      // BN+GELU of previous layer on load
        a0 = gelu_exact(a0 * sScale[k + khalf + 0] + sShift[k + khalf + 0]);
        a1 = gelu_exact(a1 * sScale[k + khalf + 1] + sShift[k + khalf + 1]);
      }
    }
    v2f a;
    a.x = a0;
    a.y = a1;
    const float* wl = sW + (k + khalf) * DD + m;              // LDS B fragments
#pragma unroll
    for (int t = 0; t < 8; ++t) {
      v2f b;
      b.x = wl[t * 16 + 0];                 // B 4x16 fp32 layout (lane=N, K pair)
      b.y = wl[t * 16 + DD];
      acc[t] = __builtin_amdgcn_wmma_f32_16x16x4_f32(
          /*neg_a=*/false, a, /*neg_b=*/false, b,
          /*c_mod=*/(short)0, acc[t], /*reuse_a=*/false, /*reuse_b=*/false);
    }
  }

  // epilogue: store Y tile + accumulate per-column BN stats
  // C/D 16x16 fp32 layout: VGPR r -> row r (lanes 0-15) / row r+8 (lanes 16-31)
  const int rowOff = (lane >> 4) * 8;
#pragma unroll
  for (int t = 0; t < 8; ++t) {
    float s = 0.0f, q = 0.0f;
#pragma unroll
    for (int r = 0; r < 8; ++r) {
      float v = acc[t][r];
      Y[(size_t)(rowBase + rowOff + r) * DD + t * 16 + m] = v;
      s += v;
      q += v * v;
    }
    atomicAdd(&lsum[t * 16 + m], s);        // ds_add_f32
    atomicAdd(&lsq[t * 16 + m], q);
  }
  __syncthreads();
  if (threadIdx.x < DD) {
    atomicAdd(&gsum[threadIdx.x], lsum[threadIdx.x]);
    atomicAdd(&gsq[threadIdx.x], lsq[threadIdx.x]);
  }
}

// ---------------------------------------------------------------------------
// 4) BatchNorm finalize + final BN+GELU apply (aggregated output)
// ---------------------------------------------------------------------------
__global__ __launch_bounds__(128) void bn_finalize_kernel(
    const float* __restrict__ sum, const float* __restrict__ sumsq,
    const float* __restrict__ g, const float* __restrict__ b,
    float* __restrict__ scale, float* __restrict__ shift)
{
  int c = threadIdx.x;
  const float invP = 1.0f / (float)PTOT;
  float m = sum[c] * invP;
  float v = sumsq[c] * invP - m * m;       // biased variance
  float is = rsqrtf(v + BNEPS);
  float sc = g[c] * is;
  scale[c] = sc;
  shift[c] = b[c] - m * sc;
}

__global__ __launch_bounds__(256) void bn_gelu_apply_kernel(
    const float* __restrict__ Y, const float* __restrict__ scale,
    const float* __restrict__ shift, float* __restrict__ out)
{
  size_t i = (size_t)blockIdx.x * 256 + threadIdx.x;
  if (i >= (size_t)PTOT * DD) return;
  int c = (int)(i & (DD - 1));
  float x = Y[i] * scale[c] + shift[c];
  out[i] = gelu_exact(x);
}

// ---------------------------------------------------------------------------
// Host-side orchestration (graph-capture safe: only async ops on `stream`)
// ---------------------------------------------------------------------------
static inline size_t align256(size_t x) { return (x + 255) & ~(size_t)255; }

extern "C" void kernel_launch(void* const* d_in, const int* in_sizes, int n_in,
                              void* d_out, int out_size, void* d_ws, size_t ws_size,
                              hipStream_t stream) {
  (void)in_sizes; (void)n_in; (void)out_size; (void)ws_size;

  const float* pts = (const float*)d_in[0];
  const float* w1  = (const float*)d_in[1];
  const float* g1  = (const float*)d_in[2];
  const float* b1  = (const float*)d_in[3];
  const float* w2  = (const float*)d_in[4];
  const float* g2  = (const float*)d_in[5];
  const float* b2  = (const float*)d_in[6];
  const float* w3  = (const float*)d_in[7];
  const float* g3  = (const float*)d_in[8];
  const float* b3  = (const float*)d_in[9];
  const float* w4  = (const float*)d_in[10];
  const float* g4  = (const float*)d_in[11];
  const float* b4  = (const float*)d_in[12];

  // ----- output regions: [aggregated P*128 | uvox.T 4*P (int bits) | norm 3*P]
  float* outAgg   = (float*)d_out;
  int*   outUvoxT = (int*)((float*)d_out + (size_t)PTOT * DD);
  float* outNorm  = (float*)d_out + (size_t)PTOT * DD + (size_t)4 * PTOT;

  // ----- workspace carve-up
  char* ws = (char*)d_ws;
  size_t o = 0;
  auto take = [&](size_t bytes) { char* p = ws + o; o += align256(bytes); return p; };
  float* YA       = (float*)take((size_t)PTOT * DD * 4);        // Y ping
  float* YB       = (float*)take((size_t)PTOT * DD * 4);        // Y pong
  float* X1       = (float*)take((size_t)PTOT * 4 * 4);         // norm pts (K=4)
  int*   keys     = (int*)  take((size_t)TS * 4);
  int*   vals     = (int*)  take((size_t)TS * 4);
  int*   counter  = (int*)  take(256);
  float* counts   = (float*)take((size_t)PTOT * 4);
  float* csum     = (float*)take((size_t)PTOT * 12);
  float* centroid = (float*)take((size_t)PTOT * 12);
  int*   uvox     = (int*)  take((size_t)PTOT * 16);
  int*   inv      = (int*)  take((size_t)PTOT * 4);
  float* w1p      = (float*)take((size_t)4 * DD * 4);
  float* w3p      = (float*)take((size_t)CONCATK * DD * 4);
  float* statSum  = (float*)take(DD * 4);
  float* statSq   = (float*)take(DD * 4);
  float* sc1 = (float*)take(DD * 4); float* sh1 = (float*)take(DD * 4);
  float* sc2 = (float*)take(DD * 4); float* sh2 = (float*)take(DD * 4);
  float* sc3 = (float*)take(DD * 4); float* sh3 = (float*)take(DD * 4);
  float* sc4 = (float*)take(DD * 4); float* sh4 = (float*)take(DD * 4);

  const int gP = PTOT / 256;            // 1024 blocks for per-point kernels
  const int gGemm = PTOT / 128;         // 2048 blocks for GEMM
  const int gElem = (PTOT * DD) / 256;  // 131072 blocks for elementwise

  // ----- voxelization -----
  (void)hipMemsetAsync(keys, 0xFF, (size_t)TS * 4, stream);      // -1 sentinel
  (void)hipMemsetAsync(vals, 0xFF, (size_t)TS * 4, stream);
  (void)hipMemsetAsync(counter, 0, 4, stream);
  (void)hipMemsetAsync(counts, 0, (size_t)PTOT * 4, stream);
  (void)hipMemsetAsync(csum, 0, (size_t)PTOT * 12, stream);
  (void)hipMemsetAsync(uvox, 0xFF, (size_t)PTOT * 16, stream);   // pad = -1

  hash_build_kernel<<<gP, 256, 0, stream>>>(pts, keys, vals, counter, counts,
                                            csum, uvox, inv);
  centroid_finalize_kernel<<<gP, 256, 0, stream>>>(counts, csum, centroid);
  post_voxel_kernel<<<gP, 256, 0, stream>>>(pts, inv, centroid, uvox,
                                            outNorm, X1, outUvoxT);
  pack_w1_kernel<<<(4 * DD + 255) / 256, 256, 0, stream>>>(w1, w1p);
  pack_w3_kernel<<<(CONCATK * DD + 255) / 256, 256, 0, stream>>>(w3, w3p);

  // ----- layer 1: (P x 4) @ (4 x 128) ; stats fused in epilogue -----
  (void)hipMemsetAsync(statSum, 0, DD * 4, stream);
  (void)hipMemsetAsync(statSq, 0, DD * 4, stream);
  gemm_wmma_f32_kernel<0><<<gGemm, 256, 0, stream>>>(
      X1, nullptr, nullptr, nullptr, nullptr, w1p, YA, statSum, statSq, 4);
  bn_finalize_kernel<<<1, 128, 0, stream>>>(statSum, statSq, g1, b1, sc1, sh1);

  // ----- layer 2: A = gelu(bn1(Y1)) fused on load -----
  (void)hipMemsetAsync(statSum, 0, DD * 4, stream);
  (void)hipMemsetAsync(statSq, 0, DD * 4, stream);
  gemm_wmma_f32_kernel<1><<<gGemm, 256, 0, stream>>>(
      YA, sc1, sh1, nullptr, nullptr, w2, YB, statSum, statSq, DD);
  bn_finalize_kernel<<<1, 128, 0, stream>>>(statSum, statSq, g2, b2, sc2, sh2);

  // ----- layer 3: virtual concat [pts | gelu(bn2(Y2)) | centroid | 0] -----
  (void)hipMemsetAsync(statSum, 0, DD * 4, stream);
  (void)hipMemsetAsync(statSq, 0, DD * 4, stream);
  gemm_wmma_f32_kernel<2><<<gGemm, 256, 0, stream>>>(
      YB, sc2, sh2, pts, centroid, w3p, YA, statSum, statSq, CONCATK);
  bn_finalize_kernel<<<1, 128, 0, stream>>>(statSum, statSq, g3, b3, sc3, sh3);

  // ----- layer 4: A = gelu(bn3(Y3)) fused on load -----
  (void)hipMemsetAsync(statSum, 0, DD * 4, stream);
  (void)hipMemsetAsync(statSq, 0, DD * 4, stream);
  gemm_wmma_f32_kernel<1><<<gGemm, 256, 0, stream>>>(
      YA, sc3, sh3, nullptr, nullptr, w4, YB, statSum, statSq, DD);
  bn_finalize_kernel<<<1, 128, 0, stream>>>(statSum, statSq, g4, b4, sc4, sh4);

  // aggregated output
  bn_gelu_apply_kernel<<<gElem, 256, 0, stream>>>(YB, sc4, sh4, outAgg);
}